// CosineSimAug_35802847379572
// MI455X (gfx1250) — compile-verified
//
#include <hip/hip_runtime.h>
#include <hip/hip_bf16.h>
#include <cstddef>

typedef float v2f __attribute__((ext_vector_type(2)));
typedef float v4f __attribute__((ext_vector_type(4)));
typedef float v8f __attribute__((ext_vector_type(8)));

#define B_  32
#define F_  256
#define N1_ 64
#define N2_ 256
#define XS  258   // transposed x-tile row stride (even -> b64-aligned B loads)

__device__ __forceinline__ v8f v8f_zero() {
  v8f z;
#pragma unroll
  for (int i = 0; i < 8; ++i) z[i] = 0.f;
  return z;
}

// D = A(16x4 f32) * B(4x16 f32) + C(16x16 f32)
__device__ __forceinline__ v8f wmma4(v2f a, v2f b, v8f c) {
  return __builtin_amdgcn_wmma_f32_16x16x4_f32(
      /*neg_a=*/false, a, /*neg_b=*/false, b,
      /*c_mod=*/(short)0, c, /*reuse_a=*/false, /*reuse_b=*/false);
}

// ---------------------------------------------------------------------------
// Kernel 1: inverse column norms. feats (B, F=256, N); inv (B, N).
// ---------------------------------------------------------------------------
__global__ __launch_bounds__(256) void colnorm_inv_kernel(
    const float* __restrict__ feats, float* __restrict__ inv, int N) {
  __shared__ float red[256];
  const int tid = threadIdx.x;
  const int c = tid & 31;
  const int wv = tid >> 5;
  const int tilesPerB = N >> 5;
  const int b = blockIdx.x / tilesPerB;
  const int col0 = (blockIdx.x % tilesPerB) * 32;

  float s = 0.f;
#pragma unroll 8
  for (int i = 0; i < 32; ++i) {
    const int f = wv + 8 * i;
    const float x = feats[((size_t)(b * F_ + f)) * N + col0 + c];
    s = fmaf(x, x, s);
  }
  red[wv * 32 + c] = s;
  __syncthreads();
  if (tid < 32) {
    float t = 0.f;
#pragma unroll
    for (int w = 0; w < 8; ++w) t += red[w * 32 + tid];
    inv[b * N + col0 + tid] = 1.f / fmaxf(sqrtf(t), 1e-8f);
  }
}

// ---------------------------------------------------------------------------
// Kernel 2: base1[b,n1,o] = b1[o] + W1[o,1:4].xyz + W1[o,4:260].tf_col
// ---------------------------------------------------------------------------
__global__ __launch_bounds__(256) void base1_kernel(
    const float* __restrict__ tf, const float* __restrict__ seeds,
    const float* __restrict__ W1g, const float* __restrict__ b1g,
    float* __restrict__ base1) {
  __shared__ float colv[256];
  __shared__ float xyzv[3];
  const int tid = threadIdx.x;
  const int b = blockIdx.x >> 6;
  const int n1 = blockIdx.x & 63;

  colv[tid] = tf[((size_t)(b * F_ + tid)) * N1_ + n1];
  if (tid < 3) xyzv[tid] = seeds[((size_t)(b * N1_ + n1)) * 3 + tid];
  __syncthreads();

  const float* wrow = W1g + (size_t)tid * 260;
  float acc = b1g[tid];
  acc = fmaf(wrow[1], xyzv[0], acc);
  acc = fmaf(wrow[2], xyzv[1], acc);
  acc = fmaf(wrow[3], xyzv[2], acc);
#pragma unroll 8
  for (int k = 0; k < 256; ++k) acc = fmaf(wrow[4 + k], colv[k], acc);

  base1[((size_t)blockIdx.x) * 256 + tid] = acc;
}

// ---------------------------------------------------------------------------
// Kernel 3 (fused heavy path): per (b, 32-col n2 tile):
//   sim tile (64 x 32), then loop over n1 PAIRS: build x1 (cols 0..31 = n1a,
//   32..63 = n1b) -> WMMA layer2 -> WMMA layer3 -> fmax regs.
// Activation tiles stored TRANSPOSED xT[col][ch] (stride 258) so each WMMA B
// operand {x[kk][n], x[kk+1][n]} is one contiguous ds_load_b64, and all
// epilogue writes are ds_store_b128.
// ---------------------------------------------------------------------------
__global__ __launch_bounds__(256) void fused_main_kernel(
    const float* __restrict__ sf, const float* __restrict__ tf,
    const float* __restrict__ W1g,
    const float* __restrict__ W2g, const float* __restrict__ b2g,
    const float* __restrict__ W3g, const float* __restrict__ b3g,
    const float* __restrict__ tinvg, const float* __restrict__ sinvg,
    const float* __restrict__ base1g, float* __restrict__ maxout) {
  extern __shared__ float smem[];
  float* reg1  = smem;            // 16640: ssub[256][32] (A) / xT[64][258] (B)
  float* tch   = smem + 16640;    // 2048 : template chunk 32f x 64n
  float* simt  = smem + 18688;    // 2112 : sim[64][33]
  float* sinvs = smem + 20800;    // 32
  float* tinvs = smem + 20832;    // 64   -> total 20896 floats = 83,584 B

  const int tid = threadIdx.x;
  const int lane = tid & 31;
  const int wv = tid >> 5;
  const int b = blockIdx.y;
  const int col0 = blockIdx.x * 32;

  if (tid < 32) sinvs[tid] = sinvg[b * N2_ + col0 + tid];
  if (tid < 64) tinvs[tid] = tinvg[b * N1_ + tid];

  const int lanelo = lane & 15;
  const int hi = lane >> 4;        // half-wave select
  const int koff = 2 * hi;         // K offset within a K=4 step
  const int rowA0 = 32 * wv + lanelo;

  // x1-build mapping: 4 consecutive channels x 16 columns per thread
  const int ch4 = (tid & 63) * 4;
  const int cc0 = (tid >> 6) * 16;
  float w1s4[4];
#pragma unroll
  for (int k = 0; k < 4; ++k) w1s4[k] = W1g[(size_t)(ch4 + k) * 260];

  // loop-invariant biases -> registers
  float bias2r[2][8], bias3r[2][8];
#pragma unroll
  for (int mt = 0; mt < 2; ++mt)
#pragma unroll
    for (int v = 0; v < 8; ++v) {
      const int ch = 32 * wv + 16 * mt + v + 8 * hi;
      bias2r[mt][v] = b2g[ch];
      bias3r[mt][v] = b3g[ch];
    }

  // ---- phase A: stage raw search tile [256f][32c] ----
#pragma unroll 8
  for (int i = 0; i < 32; ++i) {
    const int idx = tid + i * 256;
    const int f = idx >> 5, c = idx & 31;
    reg1[f * 32 + c] = sf[((size_t)(b * F_ + f)) * N2_ + col0 + c];
  }
  __syncthreads();

  // sim[n][c] = (sum_f t[f][n]*s[f][c]) * tinv[n] * sinv[c]
  float acc8[8];
#pragma unroll
  for (int j = 0; j < 8; ++j) acc8[j] = 0.f;
  const int c = lane;
  const int nb = wv;  // this wave handles n = nb*8 .. nb*8+7
  for (int kb = 0; kb < 8; ++kb) {
    __syncthreads();
#pragma unroll
    for (int i = 0; i < 8; ++i) {
      const int idx = tid + i * 256;
      const int fl = idx >> 6, n = idx & 63;
      tch[fl * 64 + n] = tf[((size_t)(b * F_ + kb * 32 + fl)) * N1_ + n];
    }
    __syncthreads();
#pragma unroll 4
    for (int fl = 0; fl < 32; ++fl) {
      const float sv = reg1[(kb * 32 + fl) * 32 + c];
#pragma unroll
      for (int j = 0; j < 8; ++j)
        acc8[j] = fmaf(tch[fl * 64 + nb * 8 + j], sv, acc8[j]);
    }
  }
#pragma unroll
  for (int j = 0; j < 8; ++j)
    simt[(nb * 8 + j) * 33 + c] = acc8[j] * tinvs[nb * 8 + j] * sinvs[c];

  // ---- phase B: layers 2&3 via WMMA, two n1 per iteration ----
  v8f vmax[2][2];
#pragma unroll
  for (int mt = 0; mt < 2; ++mt)
#pragma unroll
    for (int nt = 0; nt < 2; ++nt) vmax[mt][nt] = v8f_zero();  // ReLU => max>=0

  for (int it = 0; it < 32; ++it) {
    const int n1a = 2 * it;
    const int n1b = 2 * it + 1;
    __syncthreads();
    // build x1^T: cols cc0..cc0+15 (one n1 half), channels ch4..ch4+3
    {
      const int n1sel = (cc0 >= 32) ? n1b : n1a;
      const int sc0 = cc0 & 31;
      const v4f base4 =
          *(const v4f*)(base1g + ((size_t)(b * N1_ + n1sel)) * 256 + ch4);
#pragma unroll
      for (int j = 0; j < 16; ++j) {
        const float sv = simt[n1sel * 33 + sc0 + j];
        v4f x;
#pragma unroll
        for (int k = 0; k < 4; ++k)
          x[k] = fmaxf(fmaf(w1s4[k], sv, base4[k]), 0.f);
        *(v4f*)(&reg1[(cc0 + j) * XS + ch4]) = x;
      }
    }
    __syncthreads();

    // layer 2: acc = W2 @ x1    (2 Mtiles x 4 Ntiles per wave)
    v8f acc[2][4];
#pragma unroll
    for (int mt = 0; mt < 2; ++mt)
#pragma unroll
      for (int nt = 0; nt < 4; ++nt) acc[mt][nt] = v8f_zero();
#pragma unroll 4
    for (int k0 = 0; k0 < 256; k0 += 4) {
      const v2f a0 = *(const v2f*)(W2g + (size_t)rowA0 * 256 + k0 + koff);
      const v2f a1 = *(const v2f*)(W2g + (size_t)(rowA0 + 16) * 256 + k0 + koff);
      const int kk = k0 + koff;
      v2f bb[4];
#pragma unroll
      for (int nt = 0; nt < 4; ++nt)
        bb[nt] = *(const v2f*)(&reg1[(16 * nt + lanelo) * XS + kk]);
#pragma unroll
      for (int nt = 0; nt < 4; ++nt) {
        acc[0][nt] = wmma4(a0, bb[nt], acc[0][nt]);
        acc[1][nt] = wmma4(a1, bb[nt], acc[1][nt]);
      }
    }
    __syncthreads();  // all x1 reads done; reg1 becomes x2^T
#pragma unroll
    for (int mt = 0; mt < 2; ++mt)
#pragma unroll
      for (int nt = 0; nt < 4; ++nt) {
        const int chb = 32 * wv + 16 * mt + 8 * hi;
        const int cl = 16 * nt + lanelo;
        v4f lo, hc;
#pragma unroll
        for (int v = 0; v < 4; ++v) {
          lo[v] = fmaxf(acc[mt][nt][v] + bias2r[mt][v], 0.f);
          hc[v] = fmaxf(acc[mt][nt][v + 4] + bias2r[mt][v + 4], 0.f);
        }
        *(v4f*)(&reg1[cl * XS + chb])     = lo;
        *(v4f*)(&reg1[cl * XS + chb + 4]) = hc;
      }
    __syncthreads();

    // layer 3: acc2 = W3 @ x2, fold into running max (both n1 halves)
    v8f acc2[2][4];
#pragma unroll
    for (int mt = 0; mt < 2; ++mt)
#pragma unroll
      for (int nt = 0; nt < 4; ++nt) acc2[mt][nt] = v8f_zero();
#pragma unroll 4
    for (int k0 = 0; k0 < 256; k0 += 4) {
      const v2f a0 = *(const v2f*)(W3g + (size_t)rowA0 * 256 + k0 + koff);
      const v2f a1 = *(const v2f*)(W3g + (size_t)(rowA0 + 16) * 256 + k0 + koff);
      const int kk = k0 + koff;
      v2f bb[4];
#pragma unroll
      for (int nt = 0; nt < 4; ++nt)
        bb[nt] = *(const v2f*)(&reg1[(16 * nt + lanelo) * XS + kk]);
#pragma unroll
      for (int nt = 0; nt < 4; ++nt) {
        acc2[0][nt] = wmma4(a0, bb[nt], acc2[0][nt]);
        acc2[1][nt] = wmma4(a1, bb[nt], acc2[1][nt]);
      }
    }
#pragma unroll
    for (int mt = 0; mt < 2; ++mt)
#pragma unroll
      for (int nt = 0; nt < 4; ++nt)
#pragma unroll
        for (int v = 0; v < 8; ++v) {
          const float x3 = fmaxf(acc2[mt][nt][v] + bias3r[mt][v], 0.f);
          vmax[mt][nt & 1][v] = fmaxf(vmax[mt][nt & 1][v], x3);
        }
  }

  // store max-pooled activations (b, 256, n2)
#pragma unroll
  for (int mt = 0; mt < 2; ++mt)
#pragma unroll
    for (int nt = 0; nt < 2; ++nt)
#pragma unroll
      for (int v = 0; v < 8; ++v) {
        const int ch = 32 * wv + 16 * mt + v + 8 * hi;
        const int cl = col0 + 16 * nt + lanelo;
        maxout[((size_t)(b * 256 + ch)) * N2_ + cl] = vmax[mt][nt][v];
      }
}

// ---------------------------------------------------------------------------
// Kernel 4: head.  out = W5 @ relu(W4 @ maxin + b4) + b5 per batch.
// x4 stored transposed (stride 258) for contiguous b64 B loads in layer 5.
// ---------------------------------------------------------------------------
__global__ __launch_bounds__(256) void head_kernel(
    const float* __restrict__ maxin,
    const float* __restrict__ W4g, const float* __restrict__ b4g,
    const float* __restrict__ W5g, const float* __restrict__ b5g,
    float* __restrict__ out) {
  __shared__ __attribute__((aligned(16))) float x4[32 * XS];  // x4^T[col][ch]

  const int tid = threadIdx.x;
  const int lane = tid & 31;
  const int wv = tid >> 5;
  const int b = blockIdx.y;
  const int col0 = blockIdx.x * 32;

  const int lanelo = lane & 15;
  const int hi = lane >> 4;
  const int koff = 2 * hi;
  const int rowA0 = 32 * wv + lanelo;
  const int nA = lanelo;
  const int nB = lanelo + 16;

  float bias4r[2][8], bias5r[2][8];
#pragma unroll
  for (int mt = 0; mt < 2; ++mt)
#pragma unroll
    for (int v = 0; v < 8; ++v) {
      const int ch = 32 * wv + 16 * mt + v + 8 * hi;
      bias4r[mt][v] = b4g[ch];
      bias5r[mt][v] = b5g[ch];
    }

  // layer 4: B from global maxin (b, ch, n2)
  v8f acc[2][2];
#pragma unroll
  for (int mt = 0; mt < 2; ++mt)
#pragma unroll
    for (int nt = 0; nt < 2; ++nt) acc[mt][nt] = v8f_zero();
  const float* mb = maxin + ((size_t)b * 256) * N2_ + col0;
#pragma unroll 4
  for (int k0 = 0; k0 < 256; k0 += 4) {
    const v2f a0 = *(const v2f*)(W4g + (size_t)rowA0 * 256 + k0 + koff);
    const v2f a1 = *(const v2f*)(W4g + (size_t)(rowA0 + 16) * 256 + k0 + koff);
    const int kk = k0 + koff;
    v2f b0, b1;
    b0.x = mb[(size_t)kk * N2_ + nA];       b0.y = mb[(size_t)(kk + 1) * N2_ + nA];
    b1.x = mb[(size_t)kk * N2_ + nB];       b1.y = mb[(size_t)(kk + 1) * N2_ + nB];
    acc[0][0] = wmma4(a0, b0, acc[0][0]);
    acc[0][1] = wmma4(a0, b1, acc[0][1]);
    acc[1][0] = wmma4(a1, b0, acc[1][0]);
    acc[1][1] = wmma4(a1, b1, acc[1][1]);
  }
#pragma unroll
  for (int mt = 0; mt < 2; ++mt)
#pragma unroll
    for (int nt = 0; nt < 2; ++nt) {
      const int chb = 32 * wv + 16 * mt + 8 * hi;
      const int cl = 16 * nt + lanelo;
      v4f lo, hc;
#pragma unroll
      for (int v = 0; v < 4; ++v) {
        lo[v] = fmaxf(acc[mt][nt][v] + bias4r[mt][v], 0.f);
        hc[v] = fmaxf(acc[mt][nt][v + 4] + bias4r[mt][v + 4], 0.f);
      }
      *(v4f*)(&x4[cl * XS + chb])     = lo;
      *(v4f*)(&x4[cl * XS + chb + 4]) = hc;
    }
  __syncthreads();

  // layer 5 (no activation)
  v8f acc2[2][2];
#pragma unroll
  for (int mt = 0; mt < 2; ++mt)
#pragma unroll
    for (int nt = 0; nt < 2; ++nt) acc2[mt][nt] = v8f_zero();
#pragma unroll 4
  for (int k0 = 0; k0 < 256; k0 += 4) {
    const v2f a0 = *(const v2f*)(W5g + (size_t)rowA0 * 256 + k0 + koff);
    const v2f a1 = *(const v2f*)(W5g + (size_t)(rowA0 + 16) * 256 + k0 + koff);
    const int kk = k0 + koff;
    const v2f b0 = *(const v2f*)(&x4[nA * XS + kk]);
    const v2f b1 = *(const v2f*)(&x4[nB * XS + kk]);
    acc2[0][0] = wmma4(a0, b0, acc2[0][0]);
    acc2[0][1] = wmma4(a0, b1, acc2[0][1]);
    acc2[1][0] = wmma4(a1, b0, acc2[1][0]);
    acc2[1][1] = wmma4(a1, b1, acc2[1][1]);
  }
#pragma unroll
  for (int mt = 0; mt < 2; ++mt)
#pragma unroll
    for (int nt = 0; nt < 2; ++nt)
#pragma unroll
      for (int v = 0; v < 8; ++v) {
        const int ch = 32 * wv + 16 * mt + v + 8 * hi;
        const int cl = col0 + 16 * nt + lanelo;
        out[((size_t)(b * 256 + ch)) * N2_ + cl] = acc2[mt][nt][v] + bias5r[mt][v];
      }
}

// ---------------------------------------------------------------------------
extern "C" void kernel_launch(void* const* d_in, const int* in_sizes, int n_in,
                              void* d_out, int out_size, void* d_ws, size_t ws_size,
                              hipStream_t stream) {
  (void)in_sizes; (void)n_in; (void)out_size; (void)ws_size;
  const float* sf    = (const float*)d_in[0];   // search_feats  (32,256,256)
  const float* tf    = (const float*)d_in[1];   // template_feats(32,256,64)
  const float* seeds = (const float*)d_in[2];   // template_seeds(32,64,3)
  const float* W1 = (const float*)d_in[3];
  const float* b1 = (const float*)d_in[4];
  const float* W2 = (const float*)d_in[5];
  const float* b2 = (const float*)d_in[6];
  const float* W3 = (const float*)d_in[7];
  const float* b3 = (const float*)d_in[8];
  const float* W4 = (const float*)d_in[9];
  const float* b4 = (const float*)d_in[10];
  const float* W5 = (const float*)d_in[11];
  const float* b5 = (const float*)d_in[12];
  float* out = (float*)d_out;

  float* ws = (float*)d_ws;
  float* t_inv  = ws;                 // 2048
  float* s_inv  = ws + 2048;          // 8192
  float* base1  = ws + 10240;         // 524288
  float* maxbuf = ws + 534528;        // 2097152

  colnorm_inv_kernel<<<64, 256, 0, stream>>>(tf, t_inv, N1_);
  colnorm_inv_kernel<<<256, 256, 0, stream>>>(sf, s_inv, N2_);
  base1_kernel<<<B_ * N1_, 256, 0, stream>>>(tf, seeds, W1, b1, base1);
  fused_main_kernel<<<dim3(N2_ / 32, B_), 256, 20896 * sizeof(float), stream>>>(
      sf, tf, W1, W2, b2, W3, b3, t_inv, s_inv, base1, maxbuf);
  head_kernel<<<dim3(N2_ / 32, B_), 256, 0, stream>>>(maxbuf, W4, b4, W5, b5, out);
}